// ConvolutionBlock_75462575391265
// MI455X (gfx1250) — compile-verified
//
#include <hip/hip_runtime.h>

typedef __attribute__((ext_vector_type(2))) float v2f;
typedef __attribute__((ext_vector_type(8))) float v8f;

#define CIN  64
#define COUT 64
#define KVOL 27

// ---------------------------------------------------------------------------
// Phase 0: zero the f32 accumulator (d_out), float4 vectorized.
// ---------------------------------------------------------------------------
__global__ void zero_out(float* __restrict__ out, long total) {
  long i = ((long)blockIdx.x * blockDim.x + threadIdx.x) * 4;
  if (i + 3 < total) {
    *(float4*)(out + i) = make_float4(0.f, 0.f, 0.f, 0.f);
  } else {
    for (long j = i; j < total; ++j) out[j] = 0.f;
  }
}

// ---------------------------------------------------------------------------
// Phase 1: gather -> per-offset 16x64x64 GEMM via V_WMMA_F32_16X16X4_F32 ->
// scatter-add with native f32 atomics.
//
// Block: 256 threads = 8 wave32s. blockIdx.y = kernel offset k (0..26),
// blockIdx.x covers 8 M-tiles of 16 pairs each (128 pairs / block).
//
// LDS holds kernel[k] (64x64 f32 = 16KB) repacked in K-pairs so each lane's
// B fragment {W[k0][co], W[k0+1][co]} is a contiguous 8B ds load:
//   lds[(ci>>1)*128 + co*2 + (ci&1)] = W[ci][co]
// ---------------------------------------------------------------------------
__launch_bounds__(256)
__global__ void spconv_wmma(const float* __restrict__ feats,
                            const float* __restrict__ kern,
                            const int*   __restrict__ in_map,
                            const int*   __restrict__ out_map,
                            float*       __restrict__ out,
                            int M) {
  __shared__ __align__(16) float ldsw[32 * 128];  // 16 KB

  const int k = blockIdx.y;
  const float* kw = kern + (long)k * CIN * COUT;
  for (int e = threadIdx.x; e < CIN * COUT; e += 256) {
    int ci = e >> 6, co = e & 63;
    ldsw[(ci >> 1) * 128 + co * 2 + (ci & 1)] = kw[e];
  }
  __syncthreads();

  const int wave  = threadIdx.x >> 5;
  const int lane  = threadIdx.x & 31;
  const int m0    = (blockIdx.x * 8 + wave) * 16;
  if (m0 >= M) return;  // wave-uniform: EXEC stays all-ones for WMMA

  const int row   = lane & 15;   // M row within tile / N column within tile
  const int khalf = lane >> 4;   // 0: K{0,1}, 1: K{2,3} within each K-step

  // --- Gather A: 16 rows x 64 Cin. Lane holds {A[row][ks*4+khalf*2 + 0,1]}.
  const int   idx   = in_map[(long)k * M + m0 + row];
  const float* abase = feats + (long)idx * CIN + khalf * 2;
  v2f a[16];
#pragma unroll
  for (int ks = 0; ks < 16; ++ks)
    a[ks] = *(const v2f*)(abase + ks * 4);

  // --- Output row indices for the scatter. C/D layout: VGPR v holds
  // (M = v + khalf*8, N = row).
  int oidx[8];
#pragma unroll
  for (int v = 0; v < 8; ++v)
    oidx[v] = out_map[(long)k * M + m0 + v + khalf * 8];

  // --- 4 Cout tiles of 16x16, each = 16 chained K=4 f32 WMMAs (K total 64).
#pragma unroll
  for (int nt = 0; nt < 4; ++nt) {
    v8f c = {0.f, 0.f, 0.f, 0.f, 0.f, 0.f, 0.f, 0.f};
    const int co = nt * 16 + row;
    const float* bbase = &ldsw[khalf * 128 + co * 2];
#pragma unroll
    for (int ks = 0; ks < 16; ++ks) {
      v2f b = *(const v2f*)(bbase + ks * 256);  // {W[k0][co], W[k0+1][co]}
      c = __builtin_amdgcn_wmma_f32_16x16x4_f32(
              /*neg_a=*/false, a[ks], /*neg_b=*/false, b,
              /*c_mod=*/(short)0, c, /*reuse_a=*/false, /*reuse_b=*/false);
    }
#pragma unroll
    for (int v = 0; v < 8; ++v) {
      float* p = out + (long)oidx[v] * COUT + co;
      unsafeAtomicAdd(p, c[v]);  // native global_atomic_add_f32
    }
  }
}

// ---------------------------------------------------------------------------
// Phase 2: in-place BatchNorm(inference) + ReLU, float4 vectorized.
// out = relu(out * scale + (beta - mean*scale)), scale = gamma*rsqrt(var+eps)
// ---------------------------------------------------------------------------
__global__ void bn_relu(float* __restrict__ out,
                        const float* __restrict__ gamma,
                        const float* __restrict__ beta,
                        const float* __restrict__ rmean,
                        const float* __restrict__ rvar,
                        long total) {
  long base = ((long)blockIdx.x * blockDim.x + threadIdx.x) * 4;
  if (base >= total) return;
  float4 v = *(float4*)(out + base);
  float* vp = &v.x;
  const int c0 = (int)(base & (COUT - 1));  // COUT=64, base % 4 == 0
#pragma unroll
  for (int j = 0; j < 4; ++j) {
    const int c = c0 + j;
    const float s = gamma[c] * rsqrtf(rvar[c] + 1e-5f);
    const float b = beta[c] - rmean[c] * s;
    vp[j] = fmaxf(vp[j] * s + b, 0.f);
  }
  *(float4*)(out + base) = v;
}

// ---------------------------------------------------------------------------
extern "C" void kernel_launch(void* const* d_in, const int* in_sizes, int n_in,
                              void* d_out, int out_size, void* d_ws, size_t ws_size,
                              hipStream_t stream) {
  const float* feats   = (const float*)d_in[0];
  const float* kern    = (const float*)d_in[1];
  const float* gamma   = (const float*)d_in[2];
  const float* beta    = (const float*)d_in[3];
  const float* rmean   = (const float*)d_in[4];
  const float* rvar    = (const float*)d_in[5];
  const int*   in_map  = (const int*)d_in[6];
  const int*   out_map = (const int*)d_in[7];
  float* out = (float*)d_out;

  const int  n_points = in_sizes[0] / CIN;
  const int  M        = in_sizes[6] / KVOL;   // pairs per kernel offset
  const long total    = (long)n_points * COUT;

  // Phase 0: zero accumulator.
  {
    long nthr = (total + 3) / 4;
    int  blocks = (int)((nthr + 255) / 256);
    zero_out<<<blocks, 256, 0, stream>>>(out, total);
  }

  // Phase 1: gather -> WMMA GEMM -> atomic scatter.
  {
    dim3 grid((M + 127) / 128, KVOL);
    spconv_wmma<<<grid, 256, 0, stream>>>(feats, kern, in_map, out_map, out, M);
  }

  // Phase 2: BN + ReLU epilogue in place.
  {
    long nthr = (total + 3) / 4;
    int  blocks = (int)((nthr + 255) / 256);
    bn_relu<<<blocks, 256, 0, stream>>>(out, gamma, beta, rmean, rvar, total);
  }
}